// ESRNN_39367670235320
// MI455X (gfx1250) — compile-verified
//
#include <hip/hip_runtime.h>
#include <stdint.h>

// ESRNN / Holt-Winters recurrence, B=65536 series x T=512 steps, period M=12.
// One thread per series; double-buffered async global->LDS staging (CDNA5 TDM-style
// pipeline with s_wait_asynccnt); seasonal ring in LDS; coalesced float4 stores.

#define ROWS 256      // series per block
#define CH   16       // timesteps per staged chunk
#define NCH  32       // TT / CH
#define TT   512      // timesteps
#define MM   12       // seasonal period
#define XP   257      // padded tile stride (257 % 64 == 1 -> conflict-free)
#define SP   13       // padded seasonal ring stride (odd -> conflict-free)

#if __has_builtin(__builtin_amdgcn_global_load_async_to_lds_b32) && __has_builtin(__builtin_amdgcn_s_wait_asynccnt)
#define USE_ASYNC 1
#define WAIT_ASYNC(n) __builtin_amdgcn_s_wait_asynccnt(n)
#else
#define USE_ASYNC 0
#define WAIT_ASYNC(n)
#endif

typedef __attribute__((address_space(1))) int* gptr_i32;
typedef __attribute__((address_space(3))) int* lptr_i32;

__device__ __forceinline__ float fast_rcp(float v) {
#if __has_builtin(__builtin_amdgcn_rcpf)
  return __builtin_amdgcn_rcpf(v);   // v_rcp_f32
#else
  return 1.0f / v;
#endif
}

// Issue the 16 async loads staging one 256x16 chunk, transposed into LDS.
__device__ __forceinline__ void stage_chunk(const float* __restrict__ xblk,
                                            int t0, float* dst, int tid) {
#pragma unroll
  for (int i = 0; i < CH; ++i) {
    int flat = i * ROWS + tid;
    int r = flat >> 4;            // flat / CH : row within block
    int c = flat & (CH - 1);      // flat % CH : timestep within chunk
    const float* gp = xblk + (size_t)r * TT + t0 + c;
    float*       lp = &dst[c * XP + r];
#if USE_ASYNC
    __builtin_amdgcn_global_load_async_to_lds_b32(
        (gptr_i32)(uintptr_t)gp,
        (lptr_i32)(uint32_t)(uintptr_t)lp,
        0, 0);
#else
    *lp = *gp;
#endif
  }
}

__global__ __launch_bounds__(ROWS) void esrnn_hw_kernel(
    const float* __restrict__ x,
    const float* __restrict__ alpha_p, const float* __restrict__ beta_p,
    const float* __restrict__ phi_p,   const float* __restrict__ gamma_p,
    const float* __restrict__ l0, const float* __restrict__ b0,
    const float* __restrict__ s0, float* __restrict__ out)
{
  __shared__ float xtile[2][CH * XP];  // double-buffered [t][row] transposed tile
  __shared__ float otile[CH * XP];     // [t][row] transposed output tile
  __shared__ float sring[ROWS * SP];   // per-row 12-deep seasonal ring

  const int tid = threadIdx.x;
  const int row = blockIdx.x * ROWS + tid;

  const float alpha = alpha_p[0];
  const float beta  = beta_p[0];
  const float phi   = phi_p[0];
  const float gamma = gamma_p[0];
  const float ia = 1.0f - alpha, ib = 1.0f - beta, ig = 1.0f - gamma;

  float l = l0[row];
  float b = b0[row];
  float sf = 1.0f;                     // seasonal front, carried in a VGPR
#pragma unroll
  for (int j = 0; j < MM; ++j) {
    float v = s0[(size_t)row * MM + j];
    sring[tid * SP + j] = v;
    if (j == 0) sf = v;
  }

  const float* xblk = x   + (size_t)blockIdx.x * ROWS * TT;
  float*       oblk = out + (size_t)blockIdx.x * ROWS * TT;

  // Pipeline prologue: stage chunk 0 into buffer 0.
  stage_chunk(xblk, 0, xtile[0], tid);

  int h = 0;  // ring head: slot holding the (already-consumed-into-sf) front
  for (int chunk = 0; chunk < NCH; ++chunk) {
    const int p = chunk & 1;
    const int t0 = chunk * CH;

    // Prefetch next chunk into the other buffer, then wait for current chunk.
    if (chunk + 1 < NCH) {
      stage_chunk(xblk, t0 + CH, xtile[p ^ 1], tid);
      WAIT_ASYNC(CH);                  // older 16 (this chunk) are complete
    } else {
      WAIT_ASYNC(0);
    }
    __syncthreads();                   // everyone's tile data visible

    // ---- CH recurrence steps ----
#pragma unroll
    for (int c = 0; c < CH; ++c) {
      float xt = xtile[p][c * XP + tid];
      int h1 = (h + 1 == MM) ? 0 : h + 1;
      float ss = sring[tid * SP + h1];             // s_p[:, 1] == next front
      float bp   = b * phi;                        // b_p * phi
      float lb   = l + bp;                         // damped level forecast
      float lnew = alpha * (xt * fast_rcp(sf)) + ia * lb;
      float bnew = beta * (lnew - l) + ib * bp;
      float lbn  = lnew + bnew * phi;
      float snew = gamma * (xt * fast_rcp(lbn)) + ig * sf;
      sring[tid * SP + h] = snew;                  // push-back == overwrite front
      otile[c * XP + tid] = lbn * ss;              // out_t
      l = lnew; b = bnew; sf = ss; h = h1;
    }
    __syncthreads();                   // otile complete before cross-thread reads

    // ---- store out tile: transposed LDS -> coalesced float4 global ----
#pragma unroll
    for (int i = 0; i < CH / 4; ++i) {
      int flat4 = i * ROWS + tid;
      int r = flat4 >> 2;              // 4 float4 per (row, chunk)
      int q = flat4 & 3;
      float4 v;
      v.x = otile[(4 * q + 0) * XP + r];
      v.y = otile[(4 * q + 1) * XP + r];
      v.z = otile[(4 * q + 2) * XP + r];
      v.w = otile[(4 * q + 3) * XP + r];
      *reinterpret_cast<float4*>(oblk + (size_t)r * TT + t0 + 4 * q) = v;
    }
    // No trailing barrier needed: next iteration's pre-compute __syncthreads()
    // orders these otile reads against the next chunk's otile writes, and the
    // prefetch target buffer was last read before the post-compute barrier.
  }
}

extern "C" void kernel_launch(void* const* d_in, const int* in_sizes, int n_in,
                              void* d_out, int out_size, void* d_ws, size_t ws_size,
                              hipStream_t stream) {
  const float* x     = (const float*)d_in[0];
  const float* alpha = (const float*)d_in[1];
  const float* beta  = (const float*)d_in[2];
  const float* phi   = (const float*)d_in[3];
  const float* gamma = (const float*)d_in[4];
  const float* l0    = (const float*)d_in[5];
  const float* b0    = (const float*)d_in[6];
  const float* s0    = (const float*)d_in[7];
  float* out = (float*)d_out;

  const int B = in_sizes[5];            // 65536
  const int nblocks = B / ROWS;         // 256
  hipLaunchKernelGGL(esrnn_hw_kernel, dim3(nblocks), dim3(ROWS), 0, stream,
                     x, alpha, beta, phi, gamma, l0, b0, s0, out);
}